// LinearAttentionLayer_65197603553765
// MI455X (gfx1250) — compile-verified
//
#include <hip/hip_runtime.h>
#include <hip/hip_bf16.h>
#include <math.h>

// ---------------------------------------------------------------------------
// LinearAttentionLayer for MI455X (gfx1250, wave32, WMMA)
//   B=2, T=1024, D=256  -- all fp32, matrix math on V_WMMA_F32_16X16X4_F32.
// ---------------------------------------------------------------------------

#define D_MODEL 256
#define T_LEN   1024
#define BATCH   2
#define BT      (BATCH * T_LEN)      // 2048 rows
#define W3      (3 * D_MODEL)        // 768
#define CHUNK   16                   // time-chunk for the linear-attn scan
#define SLICE   64                   // state/value columns per workgroup
#define NSLICE  (D_MODEL / SLICE)    // 4
#define NCHUNK  (T_LEN / CHUNK)      // 64

typedef __attribute__((ext_vector_type(2))) float v2f;
typedef __attribute__((ext_vector_type(8))) float v8f;

// D = A(16x4, f32) * B(4x16, f32) + C(16x16, f32)   -- wave32 layouts:
//   A/B operand: lane l -> (row/col = l&15), k = kb + 2*(l>>4) + j, j in {0,1}
//   C/D:         vgpr r, lane l -> row = r + 8*(l>>4), col = l&15
__device__ __forceinline__ v8f wmma4(v2f a, v2f b, v8f c) {
  return __builtin_amdgcn_wmma_f32_16x16x4_f32(
      /*neg_a=*/false, a, /*neg_b=*/false, b,
      /*c_mod=*/(short)0, c, /*reuse_a=*/false, /*reuse_b=*/false);
}

__device__ __forceinline__ float softplus_f(float u) {
  return (u > 20.f) ? u : log1pf(__expf(u));
}

// ---------------------------------------------------------------------------
// Kernel 1: qkv = x @ w_qkv + b_qkv ; q,k <- softplus(l2norm(q/sqrt(D)))+1e-6
//   grid = BT/16 = 128 blocks, 512 threads (16 waves). Each block: 16 rows.
//   Each wave computes 3 of the 48 column tiles (16x16), K=256 in steps of 4.
// ---------------------------------------------------------------------------
__global__ __launch_bounds__(512)
void qkv_kernel(const float* __restrict__ x, const float* __restrict__ wq,
                const float* __restrict__ bq,
                float* __restrict__ qo, float* __restrict__ ko,
                float* __restrict__ vo) {
  __shared__ float xbuf[CHUNK * D_MODEL];       // 16 KB : A rows
  __shared__ float qkbuf[CHUNK * 2 * D_MODEL];  // 32 KB : q|k staging for norms
  __shared__ float rnorm[2 * CHUNK];

  const int row0 = blockIdx.x * CHUNK;
  const int tid  = threadIdx.x;
  const int lane = tid & 31;
  const int wave = tid >> 5;
  const int l16  = lane & 15;
  const int lh   = lane >> 4;

  for (int i = tid; i < CHUNK * D_MODEL; i += 512)
    xbuf[i] = x[(size_t)row0 * D_MODEL + i];
  __syncthreads();

#pragma unroll
  for (int t = 0; t < 3; ++t) {
    const int nt   = wave * 3 + t;   // 0..47 column tiles of 768
    const int col0 = nt * 16;
    v8f acc = {};
    for (int kb = 0; kb < D_MODEL; kb += 4) {
      const int ka = kb + 2 * lh;
      v2f a;  a.x  = xbuf[l16 * D_MODEL + ka];
              a.y  = xbuf[l16 * D_MODEL + ka + 1];
      v2f bb; bb.x = wq[(size_t)ka * W3 + col0 + l16];
              bb.y = wq[(size_t)(ka + 1) * W3 + col0 + l16];
      acc = wmma4(a, bb, acc);
    }
    const float bias = bq[col0 + l16];
#pragma unroll
    for (int r = 0; r < 8; ++r) {
      const int lrow = r + 8 * lh;
      const float val = acc[r] + bias;
      if (col0 < 2 * D_MODEL)  // q or k -> stage for row-wise normalize
        qkbuf[lrow * (2 * D_MODEL) + col0 + l16] = val;
      else                     // v -> straight to global
        vo[(size_t)(row0 + lrow) * D_MODEL + (col0 - 2 * D_MODEL) + l16] = val;
    }
  }
  __syncthreads();

  // 32 row-norm tasks (16 q-rows + 16 k-rows); wave w does tasks w and w+16.
  // normalize(q/16) == q / max(||q||, 16*1e-12)  (the 1/16 scale cancels)
  for (int task = wave; task < 32; task += 16) {
    const int lrow = task & 15, which = task >> 4;
    const float* p = qkbuf + lrow * (2 * D_MODEL) + which * D_MODEL;
    float s = 0.f;
    for (int c = lane; c < D_MODEL; c += 32) { const float u = p[c]; s += u * u; }
    for (int off = 16; off; off >>= 1) s += __shfl_xor(s, off, 32);
    if (lane == 0) rnorm[which * 16 + lrow] = 1.f / fmaxf(sqrtf(s), 1.6e-11f);
  }
  __syncthreads();

  for (int i = tid; i < CHUNK * 2 * D_MODEL; i += 512) {
    const int lrow = i >> 9, c = i & 511, which = c >> 8, d = c & 255;
    const float u = qkbuf[i] * rnorm[which * 16 + lrow];
    float* dst = which ? ko : qo;
    dst[(size_t)(row0 + lrow) * D_MODEL + d] = softplus_f(u) + 1e-6f;
  }
}

// ---------------------------------------------------------------------------
// Kernel 2: chunked causal linear attention, register-double-buffered.
//   State columns are independent -> grid = B * NSLICE = 8 workgroups; each
//   owns a 256 x 64 slice of the running state S in LDS and scans T in
//   16-row chunks:
//     Aqk       = Q Kt          (WMMA, K=256)   -> causal mask, rowsum = den
//     num_inter = Q  @ S        (WMMA, K=256)
//     num_intra = mask(Aqk) @ V (WMMA, K=16)
//     S        += Kt @ V        (WMMA, 64 tiles / 16 waves)
//   Chunk c+1's q/k/v are fetched into registers while chunk c computes, so
//   global latency is off the scan's critical path (only 8 WGs resident --
//   there is no other occupancy to hide it).  ~105 KB dynamic LDS / WG.
// ---------------------------------------------------------------------------
#define ATTN_SMEM_FLOATS (D_MODEL * SLICE + 2 * CHUNK * D_MODEL + CHUNK * SLICE \
                          + CHUNK * CHUNK + D_MODEL + 2 * CHUNK)
#define ATTN_SMEM_BYTES  (ATTN_SMEM_FLOATS * sizeof(float))

__global__ __launch_bounds__(512)
void attn_kernel(const float* __restrict__ q, const float* __restrict__ k,
                 const float* __restrict__ v, float* __restrict__ attn) {
  extern __shared__ float smem[];
  float* S    = smem;                       // [256][64]  running state slice
  float* qbuf = S + D_MODEL * SLICE;        // [16][256]
  float* kbuf = qbuf + CHUNK * D_MODEL;     // [16][256]
  float* vbuf = kbuf + CHUNK * D_MODEL;     // [16][64]
  float* aqk  = vbuf + CHUNK * SLICE;       // [16][16]  masked Q Kt
  float* ksum = aqk + CHUNK * CHUNK;        // [256]     running sum of k
  float* qdk  = ksum + D_MODEL;             // [16]      q . ksum_prev
  float* den  = qdk + CHUNK;                // [16]

  const int b     = blockIdx.x >> 2;
  const int slice = blockIdx.x & 3;
  const int tid   = threadIdx.x;
  const int lane  = tid & 31;
  const int wave  = tid >> 5;
  const int l16   = lane & 15;
  const int lh    = lane >> 4;

  for (int i = tid; i < D_MODEL * SLICE; i += 512) S[i] = 0.f;
  for (int i = tid; i < D_MODEL; i += 512) ksum[i] = 0.f;

  const size_t rowbase = (size_t)b * T_LEN * D_MODEL;

  // -- per-thread staging registers for the *next* chunk -------------------
  float rq[8], rk[8], rv[2];
#define LOAD_CHUNK(c)                                                         \
  do {                                                                        \
    const size_t base = rowbase + (size_t)(c) * CHUNK * D_MODEL;              \
    _Pragma("unroll") for (int j = 0; j < 8; ++j) {                           \
      rq[j] = q[base + tid + j * 512];                                        \
      rk[j] = k[base + tid + j * 512];                                        \
    }                                                                         \
    _Pragma("unroll") for (int j = 0; j < 2; ++j) {                           \
      const int i = tid + j * 512;                                            \
      rv[j] = v[rowbase + (size_t)((c) * CHUNK + (i >> 6)) * D_MODEL +        \
                slice * SLICE + (i & 63)];                                    \
    }                                                                         \
  } while (0)

  LOAD_CHUNK(0);
  __syncthreads();  // covers the S/ksum zero-fill as well

  for (int c = 0; c < NCHUNK; ++c) {
    const int t0 = c * CHUNK;

    // ---- publish staged chunk to LDS -------------------------------------
#pragma unroll
    for (int j = 0; j < 8; ++j) {
      qbuf[tid + j * 512] = rq[j];
      kbuf[tid + j * 512] = rk[j];
    }
#pragma unroll
    for (int j = 0; j < 2; ++j) vbuf[tid + j * 512] = rv[j];
    __syncthreads();

    // ---- kick off next chunk's global loads (overlap with compute) -------
    if (c + 1 < NCHUNK) LOAD_CHUNK(c + 1);

    // ---- qdk[w] = dot(q[row w], ksum_prev)  (one wave per row) -----------
    {
      float s = 0.f;
      const float* qp = qbuf + wave * D_MODEL;
      for (int d = lane; d < D_MODEL; d += 32) s += qp[d] * ksum[d];
      for (int off = 16; off; off >>= 1) s += __shfl_xor(s, off, 32);
      if (lane == 0) qdk[wave] = s;
    }
    __syncthreads();

    // ---- phase B1: read-only on S ----------------------------------------
    v8f numacc = {};  // live across the barrier for waves 0..3
    if (wave < 4) {
      // num_inter tile: output cols [wave*16, wave*16+16) of the slice
      const int col0 = wave * 16;
      for (int kb = 0; kb < D_MODEL; kb += 4) {
        const int ka = kb + 2 * lh;
        v2f a;  a.x  = qbuf[l16 * D_MODEL + ka];
                a.y  = qbuf[l16 * D_MODEL + ka + 1];
        v2f bb; bb.x = S[ka * SLICE + col0 + l16];
                bb.y = S[(ka + 1) * SLICE + col0 + l16];
        numacc = wmma4(a, bb, numacc);
      }
    } else if (wave == 15) {
      // Aqk = Q Kt  (Kt read as B-operand: B[d][s] = kbuf[s][d])
      v8f acc = {};
      for (int kb = 0; kb < D_MODEL; kb += 4) {
        const int ka = kb + 2 * lh;
        v2f a;  a.x  = qbuf[l16 * D_MODEL + ka];
                a.y  = qbuf[l16 * D_MODEL + ka + 1];
        v2f bb; bb.x = kbuf[l16 * D_MODEL + ka];
                bb.y = kbuf[l16 * D_MODEL + ka + 1];
        acc = wmma4(a, bb, acc);
      }
#pragma unroll
      for (int r = 0; r < 8; ++r) {   // causal mask + rowsum -> denominator
        const int row = r + 8 * lh;
        const float m = (l16 <= row) ? acc[r] : 0.f;
        aqk[row * 16 + l16] = m;
        float s = m;
        for (int off = 1; off < 16; off <<= 1) s += __shfl_xor(s, off, 32);
        if (l16 == 0) den[row] = s + qdk[row];
      }
    } else if (wave >= 4 && wave < 12) {
      // fold chunk into ksum (256 lanes <-> 256 features); read by the NEXT
      // chunk's qdk phase only, so no conflict with this chunk's consumers.
      const int d = tid - 4 * 32;   // 0..255
      float s = ksum[d];
#pragma unroll
      for (int r = 0; r < CHUNK; ++r) s += kbuf[r * D_MODEL + d];
      ksum[d] = s;
    }
    __syncthreads();

    // ---- phase B2: finish numerator, write out, then update S ------------
    if (wave < 4) {
      const int col0 = wave * 16;
      for (int kb = 0; kb < CHUNK; kb += 4) {  // num_intra: mask(Aqk) @ V
        const int ka = kb + 2 * lh;
        v2f a;  a.x  = aqk[l16 * 16 + ka];
                a.y  = aqk[l16 * 16 + ka + 1];
        v2f bb; bb.x = vbuf[ka * SLICE + col0 + l16];
                bb.y = vbuf[(ka + 1) * SLICE + col0 + l16];
        numacc = wmma4(a, bb, numacc);
      }
#pragma unroll
      for (int r = 0; r < 8; ++r) {
        const int row = r + 8 * lh;
        const float nm = fminf(fmaxf(numacc[r], -10000.f), 10000.f);
        const float dd = fmaxf(den[row], 1e-6f);
        attn[rowbase + (size_t)(t0 + row) * D_MODEL + slice * SLICE + col0 + l16] =
            nm / dd;
      }
    }
    {
      // S += Kt @ V : wave w owns row-tile w (rows w*16..), all 4 col-tiles
      const int m0 = wave * 16;
#pragma unroll
      for (int ct = 0; ct < 4; ++ct) {
        const int col0 = ct * 16;
        v8f acc;
#pragma unroll
        for (int r = 0; r < 8; ++r)
          acc[r] = S[(m0 + r + 8 * lh) * SLICE + col0 + l16];
#pragma unroll
        for (int kb = 0; kb < CHUNK; kb += 4) {
          const int ka = kb + 2 * lh;
          v2f a;  a.x  = kbuf[ka * D_MODEL + m0 + l16];        // Kt[m][t]
                  a.y  = kbuf[(ka + 1) * D_MODEL + m0 + l16];
          v2f bb; bb.x = vbuf[ka * SLICE + col0 + l16];
                  bb.y = vbuf[(ka + 1) * SLICE + col0 + l16];
          acc = wmma4(a, bb, acc);
        }
#pragma unroll
        for (int r = 0; r < 8; ++r)
          S[(m0 + r + 8 * lh) * SLICE + col0 + l16] = acc[r];
      }
    }
    __syncthreads();  // buffers free for next chunk's publish
  }
#undef LOAD_CHUNK
}

// ---------------------------------------------------------------------------
// Kernel 3: y = x + attn @ w_proj + b_proj ; out = LayerNorm(y)*gamma + beta
//   grid = 128 blocks, 512 threads; block-row GEMM + fused per-row LN.
// ---------------------------------------------------------------------------
__global__ __launch_bounds__(512)
void proj_ln_kernel(const float* __restrict__ attn, const float* __restrict__ x,
                    const float* __restrict__ wp, const float* __restrict__ bp,
                    const float* __restrict__ gamma, const float* __restrict__ beta,
                    float* __restrict__ out) {
  __shared__ float abuf[CHUNK * D_MODEL];  // 16 KB
  __shared__ float ybuf[CHUNK * D_MODEL];  // 16 KB

  const int row0 = blockIdx.x * CHUNK;
  const int tid  = threadIdx.x;
  const int lane = tid & 31;
  const int wave = tid >> 5;
  const int l16  = lane & 15;
  const int lh   = lane >> 4;

  for (int i = tid; i < CHUNK * D_MODEL; i += 512)
    abuf[i] = attn[(size_t)row0 * D_MODEL + i];
  __syncthreads();

  {
    const int col0 = wave * 16;  // 16 waves <-> 16 column tiles of 256
    v8f acc = {};
    for (int kb = 0; kb < D_MODEL; kb += 4) {
      const int ka = kb + 2 * lh;
      v2f a;  a.x  = abuf[l16 * D_MODEL + ka];
              a.y  = abuf[l16 * D_MODEL + ka + 1];
      v2f bb; bb.x = wp[(size_t)ka * D_MODEL + col0 + l16];
              bb.y = wp[(size_t)(ka + 1) * D_MODEL + col0 + l16];
      acc = wmma4(a, bb, acc);
    }
    const float bias = bp[col0 + l16];
#pragma unroll
    for (int r = 0; r < 8; ++r) {
      const int lrow = r + 8 * lh;
      ybuf[lrow * D_MODEL + col0 + l16] =
          acc[r] + bias + x[(size_t)(row0 + lrow) * D_MODEL + col0 + l16];
    }
  }
  __syncthreads();

  {  // wave w normalizes row w
    const float* yp = ybuf + wave * D_MODEL;
    float s1 = 0.f, s2 = 0.f;
    for (int c = lane; c < D_MODEL; c += 32) {
      const float u = yp[c];
      s1 += u; s2 += u * u;
    }
    for (int off = 16; off; off >>= 1) {
      s1 += __shfl_xor(s1, off, 32);
      s2 += __shfl_xor(s2, off, 32);
    }
    const float mu   = s1 * (1.f / D_MODEL);
    const float var  = s2 * (1.f / D_MODEL) - mu * mu;
    const float rstd = rsqrtf(var + 1e-5f);
    for (int c = lane; c < D_MODEL; c += 32)
      out[(size_t)(row0 + wave) * D_MODEL + c] =
          (yp[c] - mu) * rstd * gamma[c] + beta[c];
  }
}

// ---------------------------------------------------------------------------
extern "C" void kernel_launch(void* const* d_in, const int* in_sizes, int n_in,
                              void* d_out, int out_size, void* d_ws, size_t ws_size,
                              hipStream_t stream) {
  const float* x      = (const float*)d_in[0];
  const float* w_qkv  = (const float*)d_in[1];
  const float* b_qkv  = (const float*)d_in[2];
  const float* w_proj = (const float*)d_in[3];
  const float* b_proj = (const float*)d_in[4];
  const float* gamma  = (const float*)d_in[5];
  const float* beta   = (const float*)d_in[6];
  float* out = (float*)d_out;

  // workspace: q | k | v | attn, each [BT, D] fp32 (8 MB total); every element
  // is rewritten before being read each call -> deterministic.
  float* qbuf = (float*)d_ws;
  float* kbuf = qbuf + (size_t)BT * D_MODEL;
  float* vbuf = kbuf + (size_t)BT * D_MODEL;
  float* attn = vbuf + (size_t)BT * D_MODEL;

  // attn_kernel needs ~105 KB dynamic LDS (> default opt-in threshold)
  (void)hipFuncSetAttribute((const void*)attn_kernel,
                            hipFuncAttributeMaxDynamicSharedMemorySize,
                            (int)ATTN_SMEM_BYTES);

  qkv_kernel<<<BT / CHUNK, 512, 0, stream>>>(x, w_qkv, b_qkv, qbuf, kbuf, vbuf);
  attn_kernel<<<BATCH * NSLICE, 512, ATTN_SMEM_BYTES, stream>>>(qbuf, kbuf, vbuf, attn);
  proj_ln_kernel<<<BT / CHUNK, 512, 0, stream>>>(attn, x, w_proj, b_proj,
                                                 gamma, beta, out);
}